// LeakyRingAttractor_90039694393959
// MI455X (gfx1250) — compile-verified
//
#include <hip/hip_runtime.h>
#include <math.h>

#define Nn 512
#define Bb 64
#define Tt 512
#define Hh 16

typedef float v2f __attribute__((ext_vector_type(2)));
typedef float v8f __attribute__((ext_vector_type(8)));

// ---------------------------------------------------------------------------
// Pack kernel: build WMMA-B fragments for the combined weight matrix.
// Combined B (K=512 x Ncols): col-tiles c:
//   c in [0,32)  : J1*Wo.T + j0   (pre_o part, j0*sum(r) folded in)
//   c in [32,64) : Wl.T
//   c in [64,96) : Wr.T
//   c in [96,128): W_delta7 (cos(2*pi*(k-j)/N))
// Fragment layout (one float4 per lane covers two K=4 WMMA tiles, K=8 total):
//   idx = (((c*64 + kq)*32 + lane)*4 + f)
//   k   = 8*kq + (f>=2 ? 4 : 0) + ((lane>>4)<<1) + (f&1)
//   j   = 16*c' + (lane&15)
// ---------------------------------------------------------------------------
__global__ void lra_pack_kernel(const float* __restrict__ Wo,
                                const float* __restrict__ Wl,
                                const float* __restrict__ Wr,
                                float* __restrict__ pack) {
  int idx = blockIdx.x * 256 + threadIdx.x;      // 128*64*32*4 = 1,048,576
  int f   = idx & 3;
  int l   = (idx >> 2) & 31;
  int kq  = (idx >> 7) & 63;
  int c   = idx >> 13;
  int k   = 8 * kq + ((f >> 1) << 2) + ((l >> 4) << 1) + (f & 1);
  int jn  = l & 15;
  float v;
  if (c < 32) {
    int j = 16 * c + jn;
    v = 0.1f * Wo[j * Nn + k] - 0.1f;            // J1*Wo.T + j0
  } else if (c < 64) {
    int j = 16 * (c - 32) + jn;
    v = Wl[j * Nn + k];
  } else if (c < 96) {
    int j = 16 * (c - 64) + jn;
    v = Wr[j * Nn + k];
  } else {
    int j = 16 * (c - 96) + jn;
    float ang = 6.283185307179586f * (float)(k - j) / (float)Nn;
    v = cosf(ang);
  }
  pack[idx] = v;
}

// ---------------------------------------------------------------------------
// Gain precompute: Lg/Rg scalar gate per (b,t).
// ---------------------------------------------------------------------------
__device__ __forceinline__ float gelu_exact(float x) {
  return 0.5f * x * (1.0f + erff(x * 0.7071067811865476f));
}
__device__ __forceinline__ float softplus_f(float x) {
  if (x > 20.0f) return x;
  if (x < -20.0f) return expf(x);
  return log1pf(expf(x));
}
__global__ void lra_gains_kernel(const float* __restrict__ av,
                                 const float* __restrict__ grw1, const float* __restrict__ grb1,
                                 const float* __restrict__ grw2, const float* __restrict__ grb2,
                                 const float* __restrict__ glw1, const float* __restrict__ glb1,
                                 const float* __restrict__ glw2, const float* __restrict__ glb2,
                                 float* __restrict__ Lg, float* __restrict__ Rg) {
  int i = blockIdx.x * 256 + threadIdx.x;        // B*T = 32768
  float a = av[i];
  float R = fmaxf(a, 0.0f);
  float L = fmaxf(-a, 0.0f);
  float sr = grb2[0], sl = glb2[0];
#pragma unroll
  for (int h = 0; h < Hh; ++h) {
    sr += grw2[h] * gelu_exact(grw1[h] * R + grb1[h]);
    sl += glw2[h] * gelu_exact(glw1[h] * L + glb1[h]);
  }
  Rg[i] = softplus_f(sr) * R;
  Lg[i] = softplus_f(sl) * L;
}

// ---------------------------------------------------------------------------
// Main recurrence kernel: one workgroup per 16-batch tile (M=16 WMMA tile),
// 512 threads = 16 waves. Wave w owns output columns [32w, 32w+32).
// Per step: GEMM1 (16x512)x(512x1536) fp32 WMMA -> pre parts, combine with
// gains + tanh leak -> r_new (LDS); GEMM2 (16x512)x(512x512) -> rd7; row-max
// via shfl_xor + cross-wave LDS reduce; normalized store.
// ---------------------------------------------------------------------------
__global__ __launch_bounds__(512, 1)
void lra_step_kernel(const float* __restrict__ pack,
                     const float* __restrict__ Lg,
                     const float* __restrict__ Rg,
                     float* __restrict__ out_rd7,
                     float* __restrict__ out_r) {
  __shared__ float rbuf[16 * Nn];        // 32 KB: current state r (16 x 512)
  __shared__ float wavemax[16 * 16];     // per-wave row maxes
  __shared__ float finalmax[16];

  const int tid  = threadIdx.x;
  const int wave = tid >> 5;             // 0..15
  const int lane = tid & 31;
  const int half = lane >> 4;            // 0/1: K-offset half within WMMA frag
  const int ln   = lane & 15;            // row (A) / column (B,C,D) index
  const int b0   = blockIdx.x * 16;

  // ---- init r0 = bump / ||bump|| broadcast across rows ----
  float nrm2 = 0.0f;
  for (int n = 0; n < Nn; ++n) {
    float ad = fabsf((float)n - 256.0f);
    float d  = fminf(ad, 512.0f - ad);
    float e  = expf(-(d * d) / (2.0f * 51.2f * 51.2f));
    nrm2 += e * e;
  }
  float inv_nrm = 1.0f / sqrtf(nrm2);
  for (int i = tid; i < 16 * Nn; i += 512) {
    int n = i & (Nn - 1);
    float ad = fabsf((float)n - 256.0f);
    float d  = fminf(ad, 512.0f - ad);
    rbuf[i] = expf(-(d * d) / (2.0f * 51.2f * 51.2f)) * inv_nrm;
  }
  __syncthreads();

  // B-fragment pointers (float4 = 2 WMMA k-tiles per lane), stride 32 per kq
  const float4* pk  = (const float4*)pack;
  const float4* bO0 = pk + (size_t)(2 * wave)       * 2048 + lane;
  const float4* bO1 = pk + (size_t)(2 * wave + 1)   * 2048 + lane;
  const float4* bL0 = pk + (size_t)(32 + 2 * wave)  * 2048 + lane;
  const float4* bL1 = pk + (size_t)(33 + 2 * wave)  * 2048 + lane;
  const float4* bR0 = pk + (size_t)(64 + 2 * wave)  * 2048 + lane;
  const float4* bR1 = pk + (size_t)(65 + 2 * wave)  * 2048 + lane;
  const float4* bD0 = pk + (size_t)(96 + 2 * wave)  * 2048 + lane;
  const float4* bD1 = pk + (size_t)(97 + 2 * wave)  * 2048 + lane;

  const float* arow = &rbuf[ln * Nn];    // A-matrix row for this lane
  const int colA = 32 * wave + ln;
  const int colB = colA + 16;

  for (int t = 0; t < Tt; ++t) {
    // ================= GEMM1: pre parts for 32 owned columns =================
    v8f aO0 = {}, aO1 = {}, aL0 = {}, aL1 = {}, aR0 = {}, aR1 = {};
    for (int kq = 0; kq < 64; ++kq) {
      v2f a0 = *(const v2f*)(arow + 8 * kq + 2 * half);        // k = 8kq+2h,+1
      v2f a1 = *(const v2f*)(arow + 8 * kq + 4 + 2 * half);    // k = 8kq+4+2h,+1
      float4 q;
      v2f x, y;
      q = bO0[kq * 32]; x = (v2f){q.x, q.y}; y = (v2f){q.z, q.w};
      aO0 = __builtin_amdgcn_wmma_f32_16x16x4_f32(false, a0, false, x, (short)0, aO0, false, false);
      aO0 = __builtin_amdgcn_wmma_f32_16x16x4_f32(false, a1, false, y, (short)0, aO0, false, false);
      q = bO1[kq * 32]; x = (v2f){q.x, q.y}; y = (v2f){q.z, q.w};
      aO1 = __builtin_amdgcn_wmma_f32_16x16x4_f32(false, a0, false, x, (short)0, aO1, false, false);
      aO1 = __builtin_amdgcn_wmma_f32_16x16x4_f32(false, a1, false, y, (short)0, aO1, false, false);
      q = bL0[kq * 32]; x = (v2f){q.x, q.y}; y = (v2f){q.z, q.w};
      aL0 = __builtin_amdgcn_wmma_f32_16x16x4_f32(false, a0, false, x, (short)0, aL0, false, false);
      aL0 = __builtin_amdgcn_wmma_f32_16x16x4_f32(false, a1, false, y, (short)0, aL0, false, false);
      q = bL1[kq * 32]; x = (v2f){q.x, q.y}; y = (v2f){q.z, q.w};
      aL1 = __builtin_amdgcn_wmma_f32_16x16x4_f32(false, a0, false, x, (short)0, aL1, false, false);
      aL1 = __builtin_amdgcn_wmma_f32_16x16x4_f32(false, a1, false, y, (short)0, aL1, false, false);
      q = bR0[kq * 32]; x = (v2f){q.x, q.y}; y = (v2f){q.z, q.w};
      aR0 = __builtin_amdgcn_wmma_f32_16x16x4_f32(false, a0, false, x, (short)0, aR0, false, false);
      aR0 = __builtin_amdgcn_wmma_f32_16x16x4_f32(false, a1, false, y, (short)0, aR0, false, false);
      q = bR1[kq * 32]; x = (v2f){q.x, q.y}; y = (v2f){q.z, q.w};
      aR1 = __builtin_amdgcn_wmma_f32_16x16x4_f32(false, a0, false, x, (short)0, aR1, false, false);
      aR1 = __builtin_amdgcn_wmma_f32_16x16x4_f32(false, a1, false, y, (short)0, aR1, false, false);
    }

    // ---- combine: pre = pre_o' + l*pre_l + r*pre_r ; r_new = 0.85 r + 0.15 tanh(pre)
    float lgv[8], rgv[8], rold0[8], rold1[8], rn0[8], rn1[8];
#pragma unroll
    for (int e = 0; e < 8; ++e) {
      int m = e + 8 * half;
      lgv[e]   = Lg[(b0 + m) * Tt + t];
      rgv[e]   = Rg[(b0 + m) * Tt + t];
      rold0[e] = rbuf[m * Nn + colA];
      rold1[e] = rbuf[m * Nn + colB];
    }
#pragma unroll
    for (int e = 0; e < 8; ++e) {
      float p0 = aO0[e] + lgv[e] * aL0[e] + rgv[e] * aR0[e];
      float p1 = aO1[e] + lgv[e] * aL1[e] + rgv[e] * aR1[e];
      rn0[e] = rold0[e] * 0.85f + 0.15f * tanhf(p0);
      rn1[e] = rold1[e] * 0.85f + 0.15f * tanhf(p1);
    }
    __syncthreads();   // all waves finished reading r_old from rbuf
#pragma unroll
    for (int e = 0; e < 8; ++e) {
      int m = e + 8 * half;
      rbuf[m * Nn + colA] = rn0[e];
      rbuf[m * Nn + colB] = rn1[e];
      out_r[((size_t)(b0 + m) * Tt + t) * Nn + colA] = rn0[e];
      out_r[((size_t)(b0 + m) * Tt + t) * Nn + colB] = rn1[e];
    }
    __syncthreads();   // rbuf now holds r_new for everyone

    // ================= GEMM2: rd7 = r_new @ W_delta7 =================
    v8f dA = {}, dB = {};
    for (int kq = 0; kq < 64; ++kq) {
      v2f a0 = *(const v2f*)(arow + 8 * kq + 2 * half);
      v2f a1 = *(const v2f*)(arow + 8 * kq + 4 + 2 * half);
      float4 q;
      v2f x, y;
      q = bD0[kq * 32]; x = (v2f){q.x, q.y}; y = (v2f){q.z, q.w};
      dA = __builtin_amdgcn_wmma_f32_16x16x4_f32(false, a0, false, x, (short)0, dA, false, false);
      dA = __builtin_amdgcn_wmma_f32_16x16x4_f32(false, a1, false, y, (short)0, dA, false, false);
      q = bD1[kq * 32]; x = (v2f){q.x, q.y}; y = (v2f){q.z, q.w};
      dB = __builtin_amdgcn_wmma_f32_16x16x4_f32(false, a0, false, x, (short)0, dB, false, false);
      dB = __builtin_amdgcn_wmma_f32_16x16x4_f32(false, a1, false, y, (short)0, dB, false, false);
    }

    // ---- row-max over N: 16-lane shuffle reduce, then cross-wave in LDS ----
    float mx[8];
#pragma unroll
    for (int e = 0; e < 8; ++e) mx[e] = fmaxf(dA[e], dB[e]);
    for (int off = 1; off < 16; off <<= 1) {
#pragma unroll
      for (int e = 0; e < 8; ++e) mx[e] = fmaxf(mx[e], __shfl_xor(mx[e], off));
    }
    if (ln == 0) {
#pragma unroll
      for (int e = 0; e < 8; ++e) wavemax[wave * 16 + e + 8 * half] = mx[e];
    }
    __syncthreads();
    if (tid < 16) {
      float fm = wavemax[tid];
      for (int w = 1; w < 16; ++w) fm = fmaxf(fm, wavemax[w * 16 + tid]);
      finalmax[tid] = fm;
    }
    __syncthreads();
#pragma unroll
    for (int e = 0; e < 8; ++e) {
      int m = e + 8 * half;
      float inv = 1.0f / finalmax[m];
      out_rd7[((size_t)(b0 + m) * Tt + t) * Nn + colA] = dA[e] * inv;
      out_rd7[((size_t)(b0 + m) * Tt + t) * Nn + colB] = dB[e] * inv;
    }
  }
}

// ---------------------------------------------------------------------------
extern "C" void kernel_launch(void* const* d_in, const int* in_sizes, int n_in,
                              void* d_out, int out_size, void* d_ws, size_t ws_size,
                              hipStream_t stream) {
  const float* av   = (const float*)d_in[0];
  const float* Wo   = (const float*)d_in[1];
  const float* Wl   = (const float*)d_in[2];
  const float* Wr   = (const float*)d_in[3];
  const float* grw1 = (const float*)d_in[4];
  const float* grb1 = (const float*)d_in[5];
  const float* grw2 = (const float*)d_in[6];
  const float* grb2 = (const float*)d_in[7];
  const float* glw1 = (const float*)d_in[8];
  const float* glb1 = (const float*)d_in[9];
  const float* glw2 = (const float*)d_in[10];
  const float* glb2 = (const float*)d_in[11];

  float* pack = (float*)d_ws;                       // 1,048,576 floats (4 MiB)
  float* LgB  = pack + (size_t)128 * 64 * 32 * 4;   // 32768 floats
  float* RgB  = LgB + (size_t)Bb * Tt;              // 32768 floats

  lra_pack_kernel<<<4096, 256, 0, stream>>>(Wo, Wl, Wr, pack);
  lra_gains_kernel<<<(Bb * Tt) / 256, 256, 0, stream>>>(
      av, grw1, grb1, grw2, grb2, glw1, glb1, glw2, glb2, LgB, RgB);

  float* out_rd7 = (float*)d_out;
  float* out_r   = out_rd7 + (size_t)Bb * Tt * Nn;
  lra_step_kernel<<<4, 512, 0, stream>>>(pack, LgB, RgB, out_rd7, out_r);
}